// LatencyPredictor_6932077216028
// MI455X (gfx1250) — compile-verified
//
#include <hip/hip_runtime.h>
#include <cstdint>
#include <cstddef>

typedef __attribute__((ext_vector_type(2))) float v2f;
typedef __attribute__((ext_vector_type(8))) float v8f;

// ---- problem constants -----------------------------------------------------
constexpr int   BATCH = 256;
constexpr int   LEN   = 4096;
constexpr int   TWIN  = 64;
constexpr int   LP    = LEN - TWIN + 1;      // 4033 windows
constexpr int   NK1   = 16;
constexpr int   NK2   = 8;

constexpr long long NL = (long long)BATCH * LP;   // 1,032,448 rows per stream
constexpr long long NA = 8 * NL;                  // d2 / w element counts

// flat output layout (per stream block), return order:
// da, ida, vala, wa, sa, series_a, first_a
constexpr long long OFF_D2    = 0;
constexpr long long OFF_IDX   = NA;
constexpr long long OFF_VAL   = NA + NL;
constexpr long long OFF_W     = NA + 2 * NL;
constexpr long long OFF_S     = 2 * NA + 2 * NL;
constexpr long long OFF_SER   = 2 * NA + 3 * NL;
constexpr long long OFF_FIRST = 2 * NA + 4 * NL;
constexpr long long S_BLOCK   = 2 * NA + 4 * NL + BATCH;  // 20,649,216 floats

// ---- kernel 1: hankel + 2-layer dendrites + softmax/STE --------------------
// grid: (ceil(LP/32), BATCH, 2 streams), block: 32 (one wave)
// Each wave: two 16-row WMMA tiles over K=64 (16 x v_wmma_f32_16x16x4_f32 each)
__global__ __launch_bounds__(32)
void dendrites_wmma_kernel(const float* __restrict__ xa, const float* __restrict__ xb,
                           const float* __restrict__ W1a, const float* __restrict__ b1a,
                           const float* __restrict__ W2a, const float* __restrict__ b2a,
                           const float* __restrict__ W1b, const float* __restrict__ b1b,
                           const float* __restrict__ W2b, const float* __restrict__ b2b,
                           float* out, float* soma_ws, int useWs)
{
    const int lid  = threadIdx.x;      // 0..31
    const int q    = lid & 15;         // M (A) / N (B,C) sub-index
    const int half = lid >> 4;         // lane-group (K split for A/B)
    const int tile = blockIdx.x;       // 32 rows of l per block
    const int b    = blockIdx.y;
    const int s    = blockIdx.z;

    const float* x  = s ? xb  : xa;
    const float* W1 = s ? W1b : W1a;
    const float* b1 = s ? b1b : b1a;
    const float* W2 = s ? W2b : W2a;
    const float* b2 = s ? b2b : b2a;

    const int l0 = tile * 32;
    // clamp rows of partial last tile (branchless -> EXEC stays all-ones)
    int r0 = l0 + q;       if (r0 > LP - 1) r0 = LP - 1;   // tile 0 row for this q
    int r1 = l0 + 16 + q;  if (r1 > LP - 1) r1 = LP - 1;   // tile 1 row

    const float* xrow = x + (size_t)b * LEN;

    // ---- layer 1 GEMM via f32 WMMA, K accumulated 4 at a time ----
    v8f c0 = {};
    v8f c1 = {};
#pragma unroll
    for (int kc = 0; kc < 16; ++kc) {
        const int t = 4 * kc + 2 * half;          // K offset held by this lane
        v2f bm;                                    // B[k][n] = W1[n][k]
        bm.x = W1[q * TWIN + t];
        bm.y = W1[q * TWIN + t + 1];
        v2f a0;                                    // A[m][k] = x[b, l+k]
        a0.x = xrow[r0 + t];
        a0.y = xrow[r0 + t + 1];
        v2f a1;
        a1.x = xrow[r1 + t];
        a1.y = xrow[r1 + t + 1];
        c0 = __builtin_amdgcn_wmma_f32_16x16x4_f32(false, a0, false, bm,
                                                   (short)0, c0, false, false);
        c1 = __builtin_amdgcn_wmma_f32_16x16x4_f32(false, a1, false, bm,
                                                   (short)0, c1, false, false);
    }

    // ---- bias + relu in C layout, stage d1 tile to LDS ----
    // C layout: lane holds column k1=q, VGPR r -> row (r + 8*half)
    __shared__ __align__(16) float lds[32 * 20];   // stride 20: pad + 16B align
    const float bias1 = b1[q];
#pragma unroll
    for (int r = 0; r < 8; ++r) {
        float v0 = c0[r] + bias1; v0 = v0 > 0.0f ? v0 : 0.0f;
        float v1 = c1[r] + bias1; v1 = v1 > 0.0f ? v1 : 0.0f;
        lds[(r + 8 * half) * 20 + q]      = v0;
        lds[(r + 8 * half + 16) * 20 + q] = v1;
    }
    __syncthreads();

    // ---- per-lane row post-processing (lane m handles row l0+m) ----
    const int  m     = lid;
    const int  l     = l0 + m;
    const bool valid = (l < LP);

    float f1[16];
    {
        const float4* row4 = (const float4*)&lds[m * 20];
        float4 fA = row4[0], fB = row4[1], fC = row4[2], fD = row4[3];
        f1[0]=fA.x; f1[1]=fA.y; f1[2]=fA.z; f1[3]=fA.w;
        f1[4]=fB.x; f1[5]=fB.y; f1[6]=fB.z; f1[7]=fB.w;
        f1[8]=fC.x; f1[9]=fC.y; f1[10]=fC.z; f1[11]=fC.w;
        f1[12]=fD.x; f1[13]=fD.y; f1[14]=fD.z; f1[15]=fD.w;
    }

    // layer 2: d2 = relu(d1 @ W2^T + b2)   (W2/b2 uniform -> scalar loads)
    float d2v[8];
#pragma unroll
    for (int j = 0; j < 8; ++j) {
        float acc = b2[j];
#pragma unroll
        for (int k = 0; k < 16; ++k) acc = fmaf(W2[j * 16 + k], f1[k], acc);
        d2v[j] = acc > 0.0f ? acc : 0.0f;
    }

    // softmax(d2 / 0.1), mean-inhibition, surrogate weights
    float z[8], e[8], soft[8], inh[8], wsur[8];
    float zm = -3.402823466e38f;
#pragma unroll
    for (int j = 0; j < 8; ++j) { z[j] = d2v[j] * 10.0f; zm = fmaxf(zm, z[j]); }
    float se = 0.0f;
#pragma unroll
    for (int j = 0; j < 8; ++j) { e[j] = __expf(z[j] - zm); se += e[j]; }
    float ssum = 0.0f;
#pragma unroll
    for (int j = 0; j < 8; ++j) { soft[j] = e[j] / se; ssum += soft[j]; }
    const float mean = ssum * 0.125f;
    float isum = 0.0f;
#pragma unroll
    for (int j = 0; j < 8; ++j) {
        float t2 = soft[j] - mean;             // INH = 1.0
        inh[j] = t2 > 0.0f ? t2 : 0.0f;
        isum += inh[j];
    }
    const float den = isum + 1e-6f;
#pragma unroll
    for (int j = 0; j < 8; ++j) wsur[j] = inh[j] / den;

    // argmax (first max), STE weights, soma input
    int   idx  = 0;
    float best = d2v[0];
#pragma unroll
    for (int j = 1; j < 8; ++j) { if (d2v[j] > best) { best = d2v[j]; idx = j; } }
    float wv[8];
    float soma = 0.0f;
#pragma unroll
    for (int j = 0; j < 8; ++j) {
        const float hard = (j == idx) ? 1.0f : 0.0f;
        wv[j] = (hard - wsur[j]) + wsur[j];    // STE forward value, fp-faithful
        soma += wv[j] * d2v[j];
    }

    if (valid) {
        const size_t sbase = (size_t)s * S_BLOCK;
        const size_t row   = (size_t)b * LP + l;
        float4* dD = (float4*)(out + sbase + OFF_D2 + row * 8);
        dD[0] = make_float4(d2v[0], d2v[1], d2v[2], d2v[3]);
        dD[1] = make_float4(d2v[4], d2v[5], d2v[6], d2v[7]);
        out[sbase + OFF_IDX + row] = (float)idx;
        out[sbase + OFF_VAL + row] = best;
        float4* dW = (float4*)(out + sbase + OFF_W + row * 8);
        dW[0] = make_float4(wv[0], wv[1], wv[2], wv[3]);
        dW[1] = make_float4(wv[4], wv[5], wv[6], wv[7]);
        if (useWs) soma_ws[((size_t)s * BATCH + b) * LP + l] = soma;
    }
}

// ---- kernel 2: LIF scan (one thread per (stream, batch) row) ---------------
__global__ __launch_bounds__(256)
void lif_scan_kernel(float* out, const float* soma_ws, int useWs)
{
    const int gid = blockIdx.x * blockDim.x + threadIdx.x;
    if (gid >= 2 * BATCH) return;
    const int s = gid / BATCH;
    const int b = gid % BATCH;

    const size_t sbase   = (size_t)s * S_BLOCK;
    const size_t rowbase = (size_t)b * LP;
    const float* srcW = out + sbase + OFF_W;
    const float* srcD = out + sbase + OFF_D2;
    float* spk = out + sbase + OFF_S;
    float* ser = out + sbase + OFF_SER;
    const float* somap = soma_ws + ((size_t)s * BATCH + b) * LP;

    float v = 0.0f, a = 0.0f, snum = 0.0f;
    int  hf = LP;
    bool hs = false;

    for (int t = 0; t < LP; ++t) {
        float I;
        if (useWs) {
            I = somap[t];
        } else {
            const float4* wv4 = (const float4*)(srcW + (rowbase + t) * 8);
            const float4* dv4 = (const float4*)(srcD + (rowbase + t) * 8);
            float4 w0 = wv4[0], w1 = wv4[1], d0 = dv4[0], d1 = dv4[1];
            I = w0.x*d0.x + w0.y*d0.y + w0.z*d0.z + w0.w*d0.w
              + w1.x*d1.x + w1.y*d1.y + w1.z*d1.z + w1.w*d1.w;
        }
        const float th    = 1.5f + 1.5f * a;                 // TH + BETA*a
        v = v + (I - v * (1.0f / 20.0f));                    // leak + input
        const float xs    = 10.0f * (v - th);                // KSLOPE
        const float ssoft = 1.0f / (1.0f + __expf(-xs));
        const float shard = (v >= th) ? 1.0f : 0.0f;
        const float sp    = (shard - ssoft) + ssoft;         // STE spike value
        snum += sp;
        const bool newly = (!hs) && (shard > 0.0f);
        if (newly) { hf = t; hs = true; }
        v = v * (1.0f - sp) + (-0.5f) * sp;                  // VRESET
        a = a + (sp - a * (1.0f / 100.0f));                  // TAU_A
        spk[rowbase + t] = sp;
        ser[rowbase + t] = snum;
    }
    out[sbase + OFF_FIRST + b] = (float)hf;
}

// ---- host launcher ---------------------------------------------------------
extern "C" void kernel_launch(void* const* d_in, const int* in_sizes, int n_in,
                              void* d_out, int out_size, void* d_ws, size_t ws_size,
                              hipStream_t stream)
{
    const float* xa  = (const float*)d_in[0];
    const float* xb  = (const float*)d_in[1];
    const float* W1a = (const float*)d_in[2];
    const float* b1a = (const float*)d_in[3];
    const float* W2a = (const float*)d_in[4];
    const float* b2a = (const float*)d_in[5];
    const float* W1b = (const float*)d_in[6];
    const float* b1b = (const float*)d_in[7];
    const float* W2b = (const float*)d_in[8];
    const float* b2b = (const float*)d_in[9];

    float* out = (float*)d_out;
    float* ws  = (float*)d_ws;
    const int useWs = (ws_size >= (size_t)2 * BATCH * LP * sizeof(float)) ? 1 : 0;

    dim3 gridA((LP + 31) / 32, BATCH, 2);     // 127 x 256 x 2 waves
    dendrites_wmma_kernel<<<gridA, 32, 0, stream>>>(
        xa, xb, W1a, b1a, W2a, b2a, W1b, b1b, W2b, b2b, out, ws, useWs);

    lif_scan_kernel<<<2, 256, 0, stream>>>(out, ws, useWs);
}